// Attention_57097295233149
// MI455X (gfx1250) — compile-verified
//
#include <hip/hip_runtime.h>
#include <math.h>

// ---------------------------------------------------------------------------
// Sizes (fixed by the reference)
// ---------------------------------------------------------------------------
#define HEADS     8
#define DIM_HEAD  64
#define DIM       512
#define N_TOK     1024
#define N_ITERS   25
#define CD_EPS    0.1f
#define CD_INVEPS 10.0f
#define CD_CONST  0.0f      // eps * log(CD_K) with CD_K = 1.0
#define LN_EPS    1e-5f
#define QK_SCALE  0.125f    // DIM_HEAD^-0.5

typedef __attribute__((ext_vector_type(2))) float v2f;
typedef __attribute__((ext_vector_type(8))) float v8f;
typedef __attribute__((ext_vector_type(4))) int   v4i;

// ---------------------------------------------------------------------------
// Kernel 1: LayerNorm over rows of 512. One block (256 thr) per token row.
// ---------------------------------------------------------------------------
__global__ void ln_kernel(const float* __restrict__ x,
                          const float* __restrict__ gamma,
                          const float* __restrict__ beta,
                          float* __restrict__ h) {
    const int row = blockIdx.x;
    const int tid = threadIdx.x;            // 256 threads, 2 elems each
    const float* xr = x + (long)row * DIM;
    __shared__ float red[256];

    float v0 = xr[tid], v1 = xr[tid + 256];
    red[tid] = v0 + v1;
    __syncthreads();
    for (int off = 128; off > 0; off >>= 1) {
        if (tid < off) red[tid] += red[tid + off];
        __syncthreads();
    }
    float mu = red[0] * (1.0f / DIM);
    __syncthreads();

    float d0 = v0 - mu, d1 = v1 - mu;
    red[tid] = d0 * d0 + d1 * d1;
    __syncthreads();
    for (int off = 128; off > 0; off >>= 1) {
        if (tid < off) red[tid] += red[tid + off];
        __syncthreads();
    }
    float rs = rsqrtf(red[0] * (1.0f / DIM) + LN_EPS);

    float* hr = h + (long)row * DIM;
    hr[tid]       = d0 * rs * gamma[tid]       + beta[tid];
    hr[tid + 256] = d1 * rs * gamma[tid + 256] + beta[tid + 256];
}

// ---------------------------------------------------------------------------
// Kernel 2: strided f32 GEMM on V_WMMA_F32_16X16X4_F32, compile-time strides.
//   C[ti*16+m, (tj+t)*16+n] (+head offsets) = alpha * sum_k A[m,k]*B[k,n]
// One wave computes NT adjacent 16x16 C tiles along N (A fragment reused).
// 4 waves / 128-thread block; blockIdx.y = head.
// Fragment layouts per CDNA5 ISA 7.12.2.
// CAUSAL: skip tile-groups with tj > ti; guard stores of the second tile.
// KLIMIT: clamp reduction at (ti+1)*16 (attn is causal-zero beyond it).
// All branches are wave-uniform => EXEC all-ones for every WMMA.
// ---------------------------------------------------------------------------
template<int SAM, int SAK, int SBK, int SBN, int SCM, int NT, bool CAUSAL, bool KLIMIT>
__global__ void __launch_bounds__(128)
wmma_gemm_f32(const float* __restrict__ A, int aHead,
              const float* __restrict__ B, int bHead,
              float* __restrict__ C, int cHead,
              int M, int N, int K, float alpha) {
    const int wave   = threadIdx.x >> 5;
    const int lane   = threadIdx.x & 31;
    const int groupsN = (N >> 4) / NT;
    const int tilesM  = M >> 4;
    const int tg = blockIdx.x * (blockDim.x >> 5) + wave;
    if (tg >= tilesM * groupsN) return;
    const int ti = tg / groupsN;
    const int tj = (tg - ti * groupsN) * NT;     // first N-tile of the group
    if (CAUSAL && tj > ti) return;               // wave-uniform exit

    const int m16  = lane & 15;
    const int half = lane >> 4;
    const int kk0  = half * 2;

    const float* Ab  = A + (long)blockIdx.y * aHead
                         + (long)(ti * 16 + m16) * SAM + kk0 * SAK;
    const float* Bb0 = B + (long)blockIdx.y * bHead
                         + (long)(tj * 16 + m16) * SBN + kk0 * SBK;

    const v8f zero = {};
    v8f acc[NT];
#pragma unroll
    for (int t = 0; t < NT; ++t) acc[t] = zero;

    int kend = K;
    if (KLIMIT) { const int kl = (ti + 1) * 16; kend = (kl < K) ? kl : K; }

#pragma unroll 8
    for (int k = 0; k < kend; k += 4) {
        v2f av;
        if (SAK == 1) {
            av = *(const v2f*)(Ab + k);                       // global_load_b64
        } else {
            av.x = Ab[(long)k * SAK];
            av.y = Ab[(long)k * SAK + SAK];
        }
#pragma unroll
        for (int t = 0; t < NT; ++t) {
            const float* Bb = Bb0 + (long)t * 16 * SBN;
            v2f bv;
            if (SBK == 1) {
                bv = *(const v2f*)(Bb + k);                   // global_load_b64
            } else {
                bv.x = Bb[(long)k * SBK];
                bv.y = Bb[(long)k * SBK + SBK];
            }
            acc[t] = __builtin_amdgcn_wmma_f32_16x16x4_f32(
                false, av, false, bv, (short)0, acc[t], false, false);
        }
    }

    const int mOff = half * 8;
#pragma unroll
    for (int t = 0; t < NT; ++t) {
        if (CAUSAL && (tj + t) > ti) break;      // wave-uniform store guard
        float* Cb = C + (long)blockIdx.y * cHead
                      + (long)(ti * 16) * SCM + ((tj + t) * 16 + m16);
#pragma unroll
        for (int r = 0; r < 8; ++r)
            Cb[(long)(r + mOff) * SCM] = acc[t][r] * alpha;
    }
}

// ---------------------------------------------------------------------------
// Kernel 3: fused coordinate descent. One block (256 thr = 8 waves) per
// (row, head). The per-row logit vector (<=4KB) is copied into LDS once
// (async global->LDS if available) and stays resident for all 25 iterations;
// the dual variable `a` is a per-row scalar. Reductions are wave32 shuffles
// + one 8-element cross-wave LDS stage (4 barriers/iteration). Finally
// writes attn = exp((s + a - relu(s+a))/eps) in place (zeros for j > i).
// ---------------------------------------------------------------------------
__global__ void __launch_bounds__(256) cd_attn_kernel(float* __restrict__ sim, int n) {
    const int row  = blockIdx.x;
    const int head = blockIdx.y;
    float* s = sim + ((long)head * n + row) * n;

    __shared__ float srow[N_TOK];
    __shared__ float wred[8];
    const int tid  = threadIdx.x;
    const int lane = tid & 31;
    const int wid  = tid >> 5;
    const int nvalid = row + 1;             // causal: j <= row

    // ---- stage the row into LDS --------------------------------------
#if __has_builtin(__builtin_amdgcn_global_load_async_to_lds_b128)
    {
        // b128 chunks; tail over-read stays inside this 1024-float row.
        const int nv4 = (nvalid + 3) >> 2;
        for (int c = tid; c < nv4; c += 256) {
            __builtin_amdgcn_global_load_async_to_lds_b128(
                (__attribute__((address_space(1))) v4i*)(const_cast<float*>(s) + 4 * c),
                (__attribute__((address_space(3))) v4i*)(srow + 4 * c),
                0, 0);
        }
#if __has_builtin(__builtin_amdgcn_s_wait_asynccnt)
        __builtin_amdgcn_s_wait_asynccnt(0);
#else
        asm volatile("s_wait_asynccnt 0x0" ::: "memory");
#endif
    }
#else
    for (int j = tid; j < nvalid; j += 256) srow[j] = s[j];
#endif
    __syncthreads();

    float a = 0.0f;                         // a_0 = 0  (b_0 = -relu(s))
    for (int t = 0; t < N_ITERS; ++t) {
        // z_j = (s_j - relu(s_j + a)) / eps ; stable LSE over valid j
        float zv[4];
        int   cnt = 0;
        float mloc = -__builtin_inff();
        for (int j = tid; j < nvalid; j += 256) {
            const float sv = srow[j];
            const float z  = (sv - fmaxf(sv + a, 0.0f)) * CD_INVEPS;
            zv[cnt++] = z;
            mloc = fmaxf(mloc, z);
        }
        // wave32 max reduce, then 8-wide cross-wave stage
#pragma unroll
        for (int off = 16; off > 0; off >>= 1)
            mloc = fmaxf(mloc, __shfl_xor(mloc, off, 32));
        if (lane == 0) wred[wid] = mloc;
        __syncthreads();
        float m = wred[0];
#pragma unroll
        for (int w = 1; w < 8; ++w) m = fmaxf(m, wred[w]);
        __syncthreads();

        float sloc = 0.0f;
        for (int c = 0; c < cnt; ++c) sloc += __expf(zv[c] - m);
#pragma unroll
        for (int off = 16; off > 0; off >>= 1)
            sloc += __shfl_xor(sloc, off, 32);
        if (lane == 0) wred[wid] = sloc;
        __syncthreads();
        float ssum = wred[0];
#pragma unroll
        for (int w = 1; w < 8; ++w) ssum += wred[w];
        __syncthreads();

        a = CD_CONST - CD_EPS * (m + __logf(ssum));
    }

    // attn row (in place over sim); masked tail -> 0
    for (int j = tid; j < n; j += 256) {
        float v = 0.0f;
        if (j < nvalid) {
            const float sv = srow[j];
            v = __expf((sv + a - fmaxf(sv + a, 0.0f)) * CD_INVEPS);
        }
        s[j] = v;
    }
}

// ---------------------------------------------------------------------------
// Host orchestration
// ---------------------------------------------------------------------------
extern "C" void kernel_launch(void* const* d_in, const int* in_sizes, int n_in,
                              void* d_out, int out_size, void* d_ws, size_t ws_size,
                              hipStream_t stream) {
    const float* x     = (const float*)d_in[0];   // [1,1024,512]
    const float* gamma = (const float*)d_in[1];   // [512]
    const float* beta  = (const float*)d_in[2];   // [512]
    const float* w_qkv = (const float*)d_in[3];   // [512,1536]
    const float* w_out = (const float*)d_in[4];   // [512,512]
    float* out = (float*)d_out;                   // [1024,512]

    float* ws  = (float*)d_ws;
    float* h   = ws;                                // 524288 f32
    float* qkv = h   + (long)N_TOK * DIM;           // 1572864 f32
    float* sim = qkv + (long)N_TOK * 3 * DIM;       // 8388608 f32 (attn in place)
    float* ctx = sim + (long)HEADS * N_TOK * N_TOK; // 524288 f32
    (void)in_sizes; (void)n_in; (void)out_size; (void)ws_size;

    const int WAVES = 4;                            // 4 waves (128 thr) / block

    // 1) LayerNorm
    ln_kernel<<<dim3(N_TOK), dim3(256), 0, stream>>>(x, gamma, beta, h);

    // 2) qkv = h[1024,512] @ w_qkv[512,1536]
    {
        const int groups = (N_TOK / 16) * (3 * DIM / 16 / 2);   // 3072
        wmma_gemm_f32<DIM, 1, 3 * DIM, 1, 3 * DIM, 2, false, false>
            <<<dim3((groups + WAVES - 1) / WAVES), dim3(32 * WAVES), 0, stream>>>(
                h, 0, w_qkv, 0, qkv, 0,
                N_TOK, 3 * DIM, DIM, 1.0f);
    }

    // 3) sim[h] = scale * Q[h] @ K[h]^T   (lower-triangle tile-groups only)
    //    Q(i,d) = qkv[i*1536 +    0 + h*64 + d]
    //    K(j,d) = qkv[j*1536 +  512 + h*64 + d]  -> B(k=d, n=j): sbK=1, sbN=1536
    {
        const int groups = (N_TOK / 16) * (N_TOK / 16 / 2);     // 2048 per head
        wmma_gemm_f32<3 * DIM, 1, 1, 3 * DIM, N_TOK, 2, true, false>
            <<<dim3((groups + WAVES - 1) / WAVES, HEADS), dim3(32 * WAVES), 0, stream>>>(
                qkv, DIM_HEAD, qkv + DIM, DIM_HEAD, sim, N_TOK * N_TOK,
                N_TOK, N_TOK, DIM_HEAD, QK_SCALE);
    }

    // 4) 25 coor-descent iterations + attn, fused per row (LDS resident)
    cd_attn_kernel<<<dim3(N_TOK, HEADS), dim3(256), 0, stream>>>(sim, N_TOK);

    // 5) ctx[i, h*64+d] = attn[h] @ V[h]   (K-loop clamped at diagonal tile)
    //    V(j,d) = qkv[j*1536 + 1024 + h*64 + d]
    {
        const int groups = (N_TOK / 16) * (DIM_HEAD / 16 / 2);  // 128 per head
        wmma_gemm_f32<N_TOK, 1, 3 * DIM, 1, DIM, 2, false, true>
            <<<dim3((groups + WAVES - 1) / WAVES, HEADS), dim3(32 * WAVES), 0, stream>>>(
                sim, N_TOK * N_TOK, qkv + 2 * DIM, DIM_HEAD, ctx, DIM_HEAD,
                N_TOK, DIM_HEAD, N_TOK, 1.0f);
    }

    // 6) out = ctx[1024,512] @ w_out[512,512]
    {
        const int groups = (N_TOK / 16) * (DIM / 16 / 2);       // 1024
        wmma_gemm_f32<DIM, 1, DIM, 1, DIM, 2, false, false>
            <<<dim3((groups + WAVES - 1) / WAVES), dim3(32 * WAVES), 0, stream>>>(
                ctx, 0, w_out, 0, out, 0,
                N_TOK, DIM, DIM, 1.0f);
    }
}